// LSTMNetwork_16801912062406
// MI455X (gfx1250) — compile-verified
//
#include <hip/hip_runtime.h>
#include <hip/hip_bf16.h>

typedef __attribute__((ext_vector_type(2))) float v2f;
typedef __attribute__((ext_vector_type(8))) float v8f;

#define TPW 8          // M-tiles (16 rows each) per wave
#define HSTRIDE 66     // LDS row stride in words (conflict-free for C-layout)

__device__ __forceinline__ float fast_tanh(float x) {
#if __has_builtin(__builtin_amdgcn_tanhf)
    return __builtin_amdgcn_tanhf(x);              // gfx1250 v_tanh_f32
#else
    float e = __expf(2.0f * x);
    return 1.0f - 2.0f / (e + 1.0f);
#endif
}

__device__ __forceinline__ float fast_sigmoid(float x) {
#if __has_builtin(__builtin_amdgcn_tanhf)
    // sigmoid(x) = 0.5*tanh(x/2) + 0.5  -> v_mul + v_tanh + v_fma (no division)
    return fmaf(0.5f, __builtin_amdgcn_tanhf(0.5f * x), 0.5f);
#elif __has_builtin(__builtin_amdgcn_rcpf)
    return __builtin_amdgcn_rcpf(1.0f + __expf(-x));   // raw v_rcp_f32
#else
    return 1.0f / (1.0f + __expf(-x));
#endif
}

// ---------------------------------------------------------------------------
// Prep: pack W gates into padded [16][256] f32, compute per-column bias
//   bias[g*64+j] = b_g[j] + sum_k prev_h[k] * U_g[k][j]   (0 for j>=60)
//   Wall[k][g*64+j] = W_g[k][j]                           (0 pad k=15, j>=60)
// ---------------------------------------------------------------------------
__global__ void lstm_prep_kernel(const float* __restrict__ prev_h,
                                 const float* __restrict__ Wf, const float* __restrict__ Uf, const float* __restrict__ bf,
                                 const float* __restrict__ Wi, const float* __restrict__ Ui, const float* __restrict__ bi,
                                 const float* __restrict__ Wc, const float* __restrict__ Uc, const float* __restrict__ bc,
                                 const float* __restrict__ Wo, const float* __restrict__ Uo, const float* __restrict__ bo,
                                 float* __restrict__ ws) {
    int tid = threadIdx.x;                 // 0..255
    int g = tid >> 6;                      // gate 0..3 (f,i,c,o)
    int j = tid & 63;                      // padded column within gate
    const float* W = (g == 0) ? Wf : (g == 1) ? Wi : (g == 2) ? Wc : Wo;
    const float* U = (g == 0) ? Uf : (g == 1) ? Ui : (g == 2) ? Uc : Uo;
    const float* B = (g == 0) ? bf : (g == 1) ? bi : (g == 2) ? bc : bo;

    float* Wall    = ws;                   // 16*256 floats
    float* biasall = ws + 16 * 256;        // 256 floats

    #pragma unroll
    for (int k = 0; k < 16; ++k) {
        float w = (k < 15 && j < 60) ? W[k * 60 + j] : 0.0f;
        Wall[k * 256 + tid] = w;
    }

    float s = 0.0f;
    if (j < 60) {
        s = B[j];
        #pragma unroll 4
        for (int k = 0; k < 60; ++k)
            s += prev_h[k] * U[k * 60 + j];
    }
    biasall[tid] = s;
}

// ---------------------------------------------------------------------------
// Main: gates via V_WMMA_F32_16X16X4_F32, activations, output projection.
// Grid: 256 blocks x 256 threads (8 waves); each wave does TPW=8 row-tiles.
// ---------------------------------------------------------------------------
__global__ __launch_bounds__(256, 1)
void lstm_main_kernel(const float* __restrict__ x,       // [B,15]
                      const float* __restrict__ prev_c,  // [60]
                      const float* __restrict__ Wy,      // [60,3]
                      const float* __restrict__ by,      // [3]
                      const float* __restrict__ ws,      // Wall + biasall
                      float* __restrict__ out) {         // [B,3]
    const float* Wall    = ws;
    const float* biasall = ws + 16 * 256;

    __shared__ float lds_h[8][16 * HSTRIDE];   // per-wave h staging

    const int lane = threadIdx.x & 31;
    const int wv   = threadIdx.x >> 5;
    const int lo   = lane & 15;     // N (or M) index within tile
    const int hi   = lane >> 4;     // lane-half select
    float* hbuf = lds_h[wv];

    // --- hoist B fragments: bfrag[kt][nt], VGPR r holds K = kt*4 + 2*hi + r
    v2f bfrag[4][16];
    #pragma unroll
    for (int kt = 0; kt < 4; ++kt) {
        const int k0 = kt * 4 + 2 * hi;
        #pragma unroll
        for (int nt = 0; nt < 16; ++nt) {
            bfrag[kt][nt].x = Wall[(k0 + 0) * 256 + nt * 16 + lo];
            bfrag[kt][nt].y = Wall[(k0 + 1) * 256 + nt * 16 + lo];
        }
    }
    // --- per-lane bias (C layout: value depends only on N = lo)
    float biasv[16];
    #pragma unroll
    for (int nt = 0; nt < 16; ++nt) biasv[nt] = biasall[nt * 16 + lo];
    // --- prev_c broadcast per column group
    float pcv[4];
    #pragma unroll
    for (int jt = 0; jt < 4; ++jt) {
        int col = jt * 16 + lo;
        pcv[jt] = (col < 60) ? prev_c[col] : 0.0f;
    }

    const int gw = blockIdx.x * 8 + wv;    // global wave index

    #pragma unroll 1
    for (int t = 0; t < TPW; ++t) {
        const int tile = gw * TPW + t;
        const int row0 = tile * 16;
        const float* xr = x + (size_t)(row0 + lo) * 15;

        // pull next tile's inputs into near caches (WGP scope)
        if (t + 1 < TPW)
            __builtin_prefetch((const void*)(x + (size_t)(row0 + 16) * 15), 0, 3);

        // --- A fragments: 16x4 f32 tile; VGPR r: K = kt*4 + 2*hi + r, M = lo
        v2f afr[4];
        #pragma unroll
        for (int kt = 0; kt < 4; ++kt) {
            const int k0 = kt * 4 + 2 * hi;            // <= 14, always valid
            afr[kt].x = xr[k0];
            // branchless K=15 pad: clamp address, zero via select (no exec branch)
            const int  k1 = k0 + 1;
            const bool vk = (k1 < 15);
            float yv = xr[vk ? k1 : 0];
            afr[kt].y = vk ? yv : 0.0f;
        }

        // --- gates per 16-column group, then activations -> h in LDS
        #pragma unroll
        for (int jt = 0; jt < 4; ++jt) {
            v8f acc[4];
            #pragma unroll
            for (int g = 0; g < 4; ++g) {
                const int nt = g * 4 + jt;
                v8f c;
                #pragma unroll
                for (int q = 0; q < 8; ++q) c[q] = biasv[nt];
                #pragma unroll
                for (int kt = 0; kt < 4; ++kt) {
                    c = __builtin_amdgcn_wmma_f32_16x16x4_f32(
                            /*neg_a=*/false, afr[kt],
                            /*neg_b=*/false, bfrag[kt][nt],
                            /*c_mod=*/(short)0, c,
                            /*reuse_a=*/false, /*reuse_b=*/false);
                }
                acc[g] = c;
            }
            const int  col   = jt * 16 + lo;
            const bool valid = (col < 60);
            #pragma unroll
            for (int q = 0; q < 8; ++q) {
                float fv = fast_sigmoid(acc[0][q]);
                float iv = fast_sigmoid(acc[1][q]);
                float cc = fast_tanh(acc[2][q]);
                float ov = fast_sigmoid(acc[3][q]);
                float cs = fv * pcv[jt] + iv * cc;
                float hv = ov * fast_tanh(cs);
                if (!valid) hv = 0.0f;
                const int m = q + 8 * hi;              // C layout row
                hbuf[m * HSTRIDE + col] = hv;
            }
        }

        // --- output projection: y[row0+m][0..2] = by + h[m,:] @ Wy
        // (DS ops are in-order within a wave; compiler inserts dscnt waits)
        if (lane < 16) {
            float y0 = by[0], y1 = by[1], y2 = by[2];
            #pragma unroll 6
            for (int k = 0; k < 60; ++k) {
                const float hk = hbuf[lane * HSTRIDE + k];
                y0 = fmaf(hk, Wy[k * 3 + 0], y0);      // Wy reads are uniform -> SMEM
                y1 = fmaf(hk, Wy[k * 3 + 1], y1);
                y2 = fmaf(hk, Wy[k * 3 + 2], y2);
            }
            float* op = out + (size_t)(row0 + lane) * 3;
            op[0] = y0; op[1] = y1; op[2] = y2;
        }
    }
}

// ---------------------------------------------------------------------------
extern "C" void kernel_launch(void* const* d_in, const int* in_sizes, int n_in,
                              void* d_out, int out_size, void* d_ws, size_t ws_size,
                              hipStream_t stream) {
    const float* x      = (const float*)d_in[0];
    const float* prev_h = (const float*)d_in[1];
    const float* prev_c = (const float*)d_in[2];
    const float* Wf = (const float*)d_in[3];
    const float* Uf = (const float*)d_in[4];
    const float* bf = (const float*)d_in[5];
    const float* Wi = (const float*)d_in[6];
    const float* Ui = (const float*)d_in[7];
    const float* bi = (const float*)d_in[8];
    const float* Wc = (const float*)d_in[9];
    const float* Uc = (const float*)d_in[10];
    const float* bc = (const float*)d_in[11];
    const float* Wo = (const float*)d_in[12];
    const float* Uo = (const float*)d_in[13];
    const float* bo = (const float*)d_in[14];
    const float* Wy = (const float*)d_in[15];
    const float* by = (const float*)d_in[16];
    float* ws = (float*)d_ws;   // 16*256 + 256 floats = 17,408 bytes
    float* out = (float*)d_out;

    lstm_prep_kernel<<<1, 256, 0, stream>>>(prev_h,
                                            Wf, Uf, bf, Wi, Ui, bi,
                                            Wc, Uc, bc, Wo, Uo, bo, ws);

    // 262144 rows / 16 = 16384 tiles; 256 blocks * 8 waves * TPW(8) = 16384
    lstm_main_kernel<<<256, 256, 0, stream>>>(x, prev_c, Wy, by, ws, out);
}